// Argphy_12240656793670
// MI455X (gfx1250) — compile-verified
//
#include <hip/hip_runtime.h>
#include <math.h>

typedef __attribute__((ext_vector_type(2))) float v2f;
typedef __attribute__((ext_vector_type(8))) float v8f;

#define HD 64
#define WD 128
#define TSAVE 201
#define NSUB 50
#define NSTEPS ((TSAVE - 1) * NSUB)
#define NY 69

// RK coefficients (fp32, matching reference)
#define cA21 0.161f
#define cA31 (-0.008480655492356989f)
#define cA32 0.335480655492357f
#define cA41 2.8971530571054935f
#define cA42 (-6.359448489975075f)
#define cA43 4.3622954328695815f
#define cA51 5.325864828439257f
#define cA52 (-11.748883564062828f)
#define cA53 7.4955393428898365f
#define cA54 (-0.09249506636175525f)
#define cA61 5.86145544294642f
#define cA62 (-12.92096931784711f)
#define cA63 8.159367898576159f
#define cA64 (-0.071584973281401f)
#define cA65 (-0.028269050394068383f)
#define cB1 0.09646076681806523f
#define cB2 0.01f
#define cB3 0.4798896504144996f
#define cB4 1.379008574103742f
#define cB5 (-3.290069515436081f)
#define cB6 2.324710524099774f

// LDS layout (float offsets). Total ~196KB (< 320KB WGP LDS).
#define OFF_W0   0        // packed W0:  8 groups * 16 chunks * 32 lanes * 2 = 8192
#define OFF_W1   8192     // packed W1:  8 * 32 * 32 * 2 = 16384
#define OFF_W2   24576    // packed W2:  16384
#define OFF_W3   40960    // packed W3:  4 * 32 * 32 * 2 = 8192
#define OFF_WHB  49152    // 64
#define OFF_Y    49216    // 72 (state+hidden, padded)
#define OFF_YT   49288    // 72
#define OFF_K    49360    // 6 * 72 = 432
#define OFF_BUF0 49792    // 128
#define OFF_BUF1 49920    // 128
#define OFF_SC   50048    // scalar scratch (beta)
#define SMEM_FLOATS 50056
#define SMEM_BYTES (SMEM_FLOATS * 4)

__device__ __forceinline__ float softplus_f(float x) {
  // numerically stable: max(x,0) + log1p(exp(-|x|))
  return fmaxf(x, 0.0f) + log1pf(expf(-fabsf(x)));
}

// Matvec slice via chained V_WMMA_F32_16X16X4_F32, all operands in LDS.
// wp_off: float offset of this wave's packed B tiles [chunk][lane][2].
// hin_off: float offset of the input vector (length 4*NCHUNK).
// A = h-chunk broadcast across all 16 rows (lanes 0-15: K0/K1, lanes 16-31: K2/K3),
// so all rows of D are identical and acc[0] = out[n0 + lane%16] in every lane.
// 4 independent accumulator chains shorten the serial WMMA dependency.
template <int NCHUNK>
__device__ __forceinline__ float wmma_slice_dot(int wp_off, int hin_off,
                                                int lane, int koff) {
  extern __shared__ float sm[];
  v8f acc0 = {}; v8f acc1 = {}; v8f acc2 = {}; v8f acc3 = {};
#pragma unroll
  for (int c = 0; c < NCHUNK; c += 4) {
    {
      v2f a; a.x = sm[hin_off + 4 * (c + 0) + koff]; a.y = sm[hin_off + 4 * (c + 0) + koff + 1];
      v2f b = *(const v2f*)&sm[wp_off + (c + 0) * 64 + lane * 2];
      acc0 = __builtin_amdgcn_wmma_f32_16x16x4_f32(false, a, false, b, (short)0, acc0, false, false);
    }
    {
      v2f a; a.x = sm[hin_off + 4 * (c + 1) + koff]; a.y = sm[hin_off + 4 * (c + 1) + koff + 1];
      v2f b = *(const v2f*)&sm[wp_off + (c + 1) * 64 + lane * 2];
      acc1 = __builtin_amdgcn_wmma_f32_16x16x4_f32(false, a, false, b, (short)0, acc1, false, false);
    }
    {
      v2f a; a.x = sm[hin_off + 4 * (c + 2) + koff]; a.y = sm[hin_off + 4 * (c + 2) + koff + 1];
      v2f b = *(const v2f*)&sm[wp_off + (c + 2) * 64 + lane * 2];
      acc2 = __builtin_amdgcn_wmma_f32_16x16x4_f32(false, a, false, b, (short)0, acc2, false, false);
    }
    {
      v2f a; a.x = sm[hin_off + 4 * (c + 3) + koff]; a.y = sm[hin_off + 4 * (c + 3) + koff + 1];
      v2f b = *(const v2f*)&sm[wp_off + (c + 3) * 64 + lane * 2];
      acc3 = __builtin_amdgcn_wmma_f32_16x16x4_f32(false, a, false, b, (short)0, acc3, false, false);
    }
  }
  return (acc0[0] + acc1[0]) + (acc2[0] + acc3[0]);
}

// One rhs(y) evaluation. Integer LDS offsets keep all addressing in
// addrspace(3) (no generic-pointer null checks on the WMMA critical path).
__device__ __attribute__((noinline)) void rhs_eval(
    int yv_off, int kv_off,
    int tid, int w, int lane, int nl, int koff,
    float bias0, float bias1, float bias2, float bias3,
    float bhb_v, float scale_v) {
  extern __shared__ float sm[];
  __syncthreads();  // y vector fully written by caller

  // beta = sigmoid(Whb @ h + bhb): wave-0 shuffle all-reduce (no extra barrier)
  if (w == 0) {
    float v = sm[OFF_WHB + lane] * sm[yv_off + 5 + lane] +
              sm[OFF_WHB + 32 + lane] * sm[yv_off + 5 + 32 + lane];
    v += __shfl_xor(v, 16, 32);
    v += __shfl_xor(v, 8, 32);
    v += __shfl_xor(v, 4, 32);
    v += __shfl_xor(v, 2, 32);
    v += __shfl_xor(v, 1, 32);
    if (lane == 0) sm[OFF_SC] = 1.0f / (1.0f + expf(-(v + bhb_v)));
  }

  // layer 0: z1 = softplus(W0 @ h + b0), 128 outs, K=64 (16 chunks)
  {
    float o = wmma_slice_dot<16>(OFF_W0 + w * 1024, yv_off + 5, lane, koff) + bias0;
    if (lane < 16) sm[OFF_BUF0 + w * 16 + nl] = softplus_f(o);
  }
  __syncthreads();
  // layer 1: z2 = softplus(W1 @ z1 + b1), K=128 (32 chunks)
  {
    float o = wmma_slice_dot<32>(OFF_W1 + w * 2048, OFF_BUF0, lane, koff) + bias1;
    if (lane < 16) sm[OFF_BUF1 + w * 16 + nl] = softplus_f(o);
  }
  __syncthreads();
  // layer 2: z3 = softplus(W2 @ z2 + b2)
  {
    float o = wmma_slice_dot<32>(OFF_W2 + w * 2048, OFF_BUF1, lane, koff) + bias2;
    if (lane < 16) sm[OFF_BUF0 + w * 16 + nl] = softplus_f(o);
  }
  __syncthreads();
  // layer 3: o = W3 @ z3 + b3 (64 outs -> waves 0..3), dh = scale * tanh(1e-4*o)
  if (w < 4) {
    float o = wmma_slice_dot<32>(OFF_W3 + w * 2048, OFF_BUF0, lane, koff) + bias3;
    float dh = scale_v * tanhf(1e-4f * o);
    if (lane < 16) sm[kv_off + 5 + w * 16 + nl] = dh;
  }
  // SEIR scalar part (thread 0 == lane 0 of wave 0, which wrote OFF_SC)
  if (tid == 0) {
    const float kk = 0.526f, aa = 0.244f, ii = 0.244f, p = 0.667f,
                f = 0.98f, ee = 0.0f, dd = 1.0f, q = 0.5f;
    float beta = sm[OFF_SC];
    float S = sm[yv_off + 0], E = sm[yv_off + 1], I = sm[yv_off + 2], Av = sm[yv_off + 3];
    float LL = ee * E + (1.0f - q) * I + dd * Av;
    float bSL = beta * S * LL;
    sm[kv_off + 0] = -bSL;
    sm[kv_off + 1] = bSL - kk * E;
    sm[kv_off + 2] = p * kk * E - aa * I;
    sm[kv_off + 3] = (1.0f - p) * kk * E - ii * Av;
    sm[kv_off + 4] = f * aa * I + ii * Av;
  }
  __syncthreads();  // k vector complete
}

__global__ void __launch_bounds__(256, 1) Argphy_ode_kernel(
    const float* __restrict__ y0, const float* __restrict__ ts,
    const float* __restrict__ W0, const float* __restrict__ b0,
    const float* __restrict__ W1, const float* __restrict__ b1,
    const float* __restrict__ W2, const float* __restrict__ b2,
    const float* __restrict__ W3, const float* __restrict__ b3,
    const float* __restrict__ Whb, const float* __restrict__ bhb,
    const float* __restrict__ hvec, const float* __restrict__ scale,
    float* __restrict__ out_state, float* __restrict__ out_hid) {
  extern __shared__ float sm[];
  const int tid = threadIdx.x;
  const int w = tid >> 5;        // wave id 0..7 (wave32)
  const int lane = tid & 31;
  const int nl = lane & 15;      // output column within 16-wide group
  const int koff = (lane < 16) ? 0 : 2;

  // ---- one-time: pack weights into per-lane WMMA-B fragment order in LDS ----
  // layout: [group][chunk][lane][2]; lane holds W[n0+lane%16, 4c + 2*(lane/16) + {0,1}]
  for (int idx = tid; idx < 8192; idx += 256) {  // W0: 128x64, 16 chunks
    int j = idx & 1, ln = (idx >> 1) & 31, c = (idx >> 6) & 15, g = idx >> 10;
    int n = g * 16 + (ln & 15);
    int k = c * 4 + 2 * (ln >> 4) + j;
    sm[OFF_W0 + idx] = W0[n * HD + k];
  }
  for (int idx = tid; idx < 16384; idx += 256) {  // W1, W2: 128x128, 32 chunks
    int j = idx & 1, ln = (idx >> 1) & 31, c = (idx >> 6) & 31, g = idx >> 11;
    int n = g * 16 + (ln & 15);
    int k = c * 4 + 2 * (ln >> 4) + j;
    sm[OFF_W1 + idx] = W1[n * WD + k];
    sm[OFF_W2 + idx] = W2[n * WD + k];
  }
  for (int idx = tid; idx < 8192; idx += 256) {   // W3: 64x128, 4 groups, 32 chunks
    int j = idx & 1, ln = (idx >> 1) & 31, c = (idx >> 6) & 31, g = idx >> 11;
    int n = g * 16 + (ln & 15);
    int k = c * 4 + 2 * (ln >> 4) + j;
    sm[OFF_W3 + idx] = W3[n * WD + k];
  }
  if (tid < HD) sm[OFF_WHB + tid] = Whb[tid];
  if (tid < 5) sm[OFF_Y + tid] = y0[tid];
  if (tid < HD) sm[OFF_Y + 5 + tid] = hvec[tid];

  const float bias0 = b0[w * 16 + nl];
  const float bias1 = b1[w * 16 + nl];
  const float bias2 = b2[w * 16 + nl];
  const float bias3 = (w < 4) ? b3[w * 16 + nl] : 0.0f;
  const float bhb_v = bhb[0];
  const float scale_v = scale[0];
  __syncthreads();

  const int k0 = OFF_K + 0 * 72, k1 = OFF_K + 1 * 72, k2 = OFF_K + 2 * 72;
  const int k3 = OFF_K + 3 * 72, k4 = OFF_K + 4 * 72, k5 = OFF_K + 5 * 72;

  for (int m = 0; m < NSTEPS; ++m) {
    if ((m % NSUB) == 0) {  // save y_all[::NSUB]
      int r = m / NSUB;
      if (tid < 5) out_state[r * 5 + tid] = sm[OFF_Y + tid];
      if (tid < HD) out_hid[r * HD + tid] = sm[OFF_Y + 5 + tid];
    }
    int iv = m / NSUB;
    float dt = (ts[iv + 1] - ts[iv]) * (1.0f / (float)NSUB);

    rhs_eval(OFF_Y, k0, tid, w, lane, nl, koff, bias0, bias1, bias2, bias3, bhb_v, scale_v);
    if (tid < NY) sm[OFF_YT + tid] = sm[OFF_Y + tid] + dt * (cA21 * sm[k0 + tid]);
    rhs_eval(OFF_YT, k1, tid, w, lane, nl, koff, bias0, bias1, bias2, bias3, bhb_v, scale_v);
    if (tid < NY) sm[OFF_YT + tid] = sm[OFF_Y + tid] + dt * (cA31 * sm[k0 + tid] + cA32 * sm[k1 + tid]);
    rhs_eval(OFF_YT, k2, tid, w, lane, nl, koff, bias0, bias1, bias2, bias3, bhb_v, scale_v);
    if (tid < NY) sm[OFF_YT + tid] = sm[OFF_Y + tid] + dt * (cA41 * sm[k0 + tid] + cA42 * sm[k1 + tid] + cA43 * sm[k2 + tid]);
    rhs_eval(OFF_YT, k3, tid, w, lane, nl, koff, bias0, bias1, bias2, bias3, bhb_v, scale_v);
    if (tid < NY) sm[OFF_YT + tid] = sm[OFF_Y + tid] + dt * (cA51 * sm[k0 + tid] + cA52 * sm[k1 + tid] + cA53 * sm[k2 + tid] + cA54 * sm[k3 + tid]);
    rhs_eval(OFF_YT, k4, tid, w, lane, nl, koff, bias0, bias1, bias2, bias3, bhb_v, scale_v);
    if (tid < NY) sm[OFF_YT + tid] = sm[OFF_Y + tid] + dt * (cA61 * sm[k0 + tid] + cA62 * sm[k1 + tid] + cA63 * sm[k2 + tid] + cA64 * sm[k3 + tid] + cA65 * sm[k4 + tid]);
    rhs_eval(OFF_YT, k5, tid, w, lane, nl, koff, bias0, bias1, bias2, bias3, bhb_v, scale_v);
    if (tid < NY)
      sm[OFF_Y + tid] = sm[OFF_Y + tid] + dt * (cB1 * sm[k0 + tid] + cB2 * sm[k1 + tid] + cB3 * sm[k2 + tid] +
                                                cB4 * sm[k3 + tid] + cB5 * sm[k4 + tid] + cB6 * sm[k5 + tid]);
    __syncthreads();  // y coherent for next iteration's save/rhs
  }
  // final saved row (r = 200)
  if (tid < 5) out_state[200 * 5 + tid] = sm[OFF_Y + tid];
  if (tid < HD) out_hid[200 * HD + tid] = sm[OFF_Y + 5 + tid];
}

extern "C" void kernel_launch(void* const* d_in, const int* in_sizes, int n_in,
                              void* d_out, int out_size, void* d_ws, size_t ws_size,
                              hipStream_t stream) {
  (void)in_sizes; (void)n_in; (void)out_size; (void)d_ws; (void)ws_size;
  const float* y0    = (const float*)d_in[0];
  const float* ts    = (const float*)d_in[1];
  const float* W0    = (const float*)d_in[2];
  const float* b0    = (const float*)d_in[3];
  const float* W1    = (const float*)d_in[4];
  const float* b1    = (const float*)d_in[5];
  const float* W2    = (const float*)d_in[6];
  const float* b2    = (const float*)d_in[7];
  const float* W3    = (const float*)d_in[8];
  const float* b3    = (const float*)d_in[9];
  const float* Whb   = (const float*)d_in[10];
  const float* bhb   = (const float*)d_in[11];
  const float* hvec  = (const float*)d_in[12];
  const float* scale = (const float*)d_in[13];
  float* out_state = (float*)d_out;
  float* out_hid   = out_state + TSAVE * 5;

  // ~196KB dynamic LDS (gfx1250 WGP supports up to 320KB)
  (void)hipFuncSetAttribute((const void*)Argphy_ode_kernel,
                            hipFuncAttributeMaxDynamicSharedMemorySize, SMEM_BYTES);
  Argphy_ode_kernel<<<dim3(1), dim3(256), SMEM_BYTES, stream>>>(
      y0, ts, W0, b0, W1, b1, W2, b2, W3, b3, Whb, bhb, hvec, scale,
      out_state, out_hid);
}